// DTWNET_SINGLE_46342697124327
// MI455X (gfx1250) — compile-verified
//
#include <hip/hip_runtime.h>
#include <hip/hip_bf16.h>

// Problem dims (compile-time)
#define N_T      24
#define NBATCH   2
#define C_IN     9
#define HH       32
#define WWD      32
#define N_FILTER 32
#define KM_      16
#define N_CLASS  10
#define HID      128
#define B2       (NBATCH * HH * WWD)   // 2048 sequences

typedef float v2f __attribute__((ext_vector_type(2)));
typedef float v8f __attribute__((ext_vector_type(8)));

__device__ __forceinline__ float fmin3f(float a, float b, float c) {
    return fminf(a, fminf(b, c));
}

// ---------------------------------------------------------------------------
// Phase 1: DTW feature bank.
// grid = 2048 blocks (one sequence each), block = 32 threads (one wave),
// lane = filter index. D matrix lives in LDS, lane-interleaved (bank-clean).
// ---------------------------------------------------------------------------
__global__ __launch_bounds__(N_FILTER)
void dtw_kernel(const float* __restrict__ x, const float* __restrict__ kern,
                float* __restrict__ feats)
{
    __shared__ float s_seq[N_T * C_IN];                 // 216 floats (shared seq)
    __shared__ float s_D[KM_ * N_T * N_FILTER];         // 16*24*32 f32 = 48 KB

    const int seq = blockIdx.x;
    const int f   = threadIdx.x;

    // seqs = x.reshape(-1, 24, 9): raw row-major view -> seq i is x_flat[i*216..]
    const float* xs = x + (size_t)seq * (N_T * C_IN);
    for (int e = f; e < N_T * C_IN; e += N_FILTER) s_seq[e] = xs[e];
    __syncthreads();

    const float FINF = __builtin_inff();
    const float* kf  = kern + (size_t)f * KM_ * C_IN;   // this lane's filter

    float prev[N_T];   // previous DP row, register-resident (loops unrolled)

    // Row 0: cumulative sum of local costs
    {
        float kr[C_IN];
#pragma unroll
        for (int c = 0; c < C_IN; ++c) kr[c] = kf[c];
        float run = 0.f;
#pragma unroll
        for (int j = 0; j < N_T; ++j) {
            float cc = 0.f;
#pragma unroll
            for (int c = 0; c < C_IN; ++c) {
                float d = kr[c] - s_seq[j * C_IN + c];
                cc += d * d;
            }
            run += cc;
            prev[j] = run;
            s_D[(0 * N_T + j) * N_FILTER + f] = run;
        }
    }

    // Rows 1..15: D[i][j] = C[i][j] + min(left, up, diag)
    for (int i = 1; i < KM_; ++i) {
        float kr[C_IN];
#pragma unroll
        for (int c = 0; c < C_IN; ++c) kr[c] = kf[i * C_IN + c];
        float left = FINF;
        float diag = FINF;          // old prev[j-1]
#pragma unroll
        for (int j = 0; j < N_T; ++j) {
            float cc = 0.f;
#pragma unroll
            for (int c = 0; c < C_IN; ++c) {
                float d = kr[c] - s_seq[j * C_IN + c];
                cc += d * d;
            }
            float up = prev[j];
            float dv = cc + fmin3f(left, up, diag);
            diag = up;              // old prev[j] is diag for column j+1
            prev[j] = dv;
            left = dv;
            s_D[(i * N_T + j) * N_FILTER + f] = dv;
        }
    }

    // Backtrack from (15,23), summing local costs along the optimal path.
    int i = KM_ - 1, j = N_T - 1;
    float acc;
    {
        float cc = 0.f;
#pragma unroll
        for (int c = 0; c < C_IN; ++c) {
            float d = kf[i * C_IN + c] - s_seq[j * C_IN + c];
            cc += d * d;
        }
        acc = cc;
    }
    for (int step = 0; step < KM_ + N_T - 2; ++step) {
        if (i == 0 && j == 0) break;
        const int im1 = i > 0 ? i - 1 : 0;
        const int jm1 = j > 0 ? j - 1 : 0;
        const float c0 = (i > 0 && j > 0) ? s_D[(im1 * N_T + jm1) * N_FILTER + f] : FINF;
        const float c1 = (i > 0)          ? s_D[(im1 * N_T + j  ) * N_FILTER + f] : FINF;
        const float c2 = (j > 0)          ? s_D[(i   * N_T + jm1) * N_FILTER + f] : FINF;
        int ni, nj;
        if (c0 <= c1 && c0 <= c2) { ni = im1; nj = jm1; }   // diag (first-min tie-break)
        else if (c1 <= c2)        { ni = im1; nj = j;   }   // up
        else                      { ni = i;   nj = jm1; }   // left
        float cc = 0.f;
#pragma unroll
        for (int c = 0; c < C_IN; ++c) {
            float d = kf[ni * C_IN + c] - s_seq[nj * C_IN + c];
            cc += d * d;
        }
        acc += cc;
        i = ni; j = nj;
    }
    feats[(size_t)seq * N_FILTER + f] = acc;
}

// ---------------------------------------------------------------------------
// Phase 2: GEMM via V_WMMA_F32_16X16X4_F32 (f32 matrix pipe, matches the f32
// reference numerics). One wave computes one 16x16 tile of Out = A*B + bias,
// optionally ReLU. 4 waves per 128-thread block.
//   A layout (16x4 f32): lane<16 -> K = k0+{0,1}; lane>=16 -> K = k0+{2,3}
//   B layout (4x16 f32): VGPR v, lane-half h -> row K = k0 + 2h + v, col = lane%16
//   D layout (16x16 f32): VGPR r, lane-half h -> row = 8h + r, col = lane%16
// ---------------------------------------------------------------------------
template<int MTILES, int NTILES, int K, int NVALID, int LDO, bool RELU>
__global__ __launch_bounds__(128)
void gemm_wmma_f32(const float* __restrict__ A, const float* __restrict__ Bm,
                   const float* __restrict__ bias, float* __restrict__ Out)
{
    constexpr bool FULL = (NVALID == NTILES * 16);
    const int wave = (int)(threadIdx.x >> 5);
    const int tile = blockIdx.x * 4 + wave;
    if (tile >= MTILES * NTILES) return;       // whole-wave uniform exit
    const int mt = tile / NTILES;
    const int nt = tile % NTILES;
    const int m0 = mt * 16, n0 = nt * 16;
    const int lane = (int)(threadIdx.x & 31);
    const int half = lane >> 4;
    const int lm   = lane & 15;
    const int n    = n0 + lm;

    v8f acc = {};
    const float* arow = A + (size_t)(m0 + lm) * K + half * 2;
    for (int k0 = 0; k0 < K; k0 += 4) {
        v2f a, b;
        a.x = arow[k0 + 0];
        a.y = arow[k0 + 1];
        const int kb = k0 + half * 2;
        if (FULL || n < NVALID) {
            b.x = Bm[(size_t)(kb + 0) * NVALID + n];
            b.y = Bm[(size_t)(kb + 1) * NVALID + n];
        } else {
            b.x = 0.f; b.y = 0.f;
        }
        acc = __builtin_amdgcn_wmma_f32_16x16x4_f32(
            /*neg_a=*/false, a, /*neg_b=*/false, b,
            /*c_mod=*/(short)0, acc, /*reuse_a=*/false, /*reuse_b=*/false);
    }

    const float bv = (FULL || n < NVALID) ? bias[n] : 0.f;
#pragma unroll
    for (int r = 0; r < 8; ++r) {
        float v = acc[r] + bv;
        if (RELU) v = fmaxf(v, 0.f);
        Out[(size_t)(m0 + half * 8 + r) * LDO + n0 + lm] = v;
    }
}

// ---------------------------------------------------------------------------
// Phase 3: row-wise softmax over 10 classes (logits padded to ld=16 in ws).
// ---------------------------------------------------------------------------
__global__ __launch_bounds__(256)
void softmax10(const float* __restrict__ logits, float* __restrict__ out)
{
    const int row = blockIdx.x * blockDim.x + threadIdx.x;
    if (row >= B2) return;
    const float* l = logits + (size_t)row * 16;
    float m = l[0];
#pragma unroll
    for (int c = 1; c < N_CLASS; ++c) m = fmaxf(m, l[c]);
    float e[N_CLASS];
    float s = 0.f;
#pragma unroll
    for (int c = 0; c < N_CLASS; ++c) { e[c] = expf(l[c] - m); s += e[c]; }
    const float inv = 1.f / s;
#pragma unroll
    for (int c = 0; c < N_CLASS; ++c)
        out[(size_t)row * N_CLASS + c] = e[c] * inv;
}

// ---------------------------------------------------------------------------
extern "C" void kernel_launch(void* const* d_in, const int* in_sizes, int n_in,
                              void* d_out, int out_size, void* d_ws, size_t ws_size,
                              hipStream_t stream)
{
    (void)in_sizes; (void)n_in; (void)out_size; (void)ws_size;
    const float* x    = (const float*)d_in[0];
    const float* kern = (const float*)d_in[1];
    const float* W1   = (const float*)d_in[2];
    const float* b1   = (const float*)d_in[3];
    const float* W2   = (const float*)d_in[4];
    const float* b2   = (const float*)d_in[5];
    const float* Wl   = (const float*)d_in[6];
    const float* bl   = (const float*)d_in[7];
    float* out = (float*)d_out;

    // Workspace layout (floats): feats | h1 | h2 | logits(pad 16)
    float* feats  = (float*)d_ws;                        // 2048*32
    float* h1     = feats + (size_t)B2 * N_FILTER;       // 2048*128
    float* h2     = h1    + (size_t)B2 * HID;            // 2048*128
    float* logits = h2    + (size_t)B2 * HID;            // 2048*16

    // 1) DTW features
    dtw_kernel<<<B2, N_FILTER, 0, stream>>>(x, kern, feats);

    // 2) MLP via f32 WMMA
    constexpr int T1 = (B2 / 16) * (HID / 16);           // 1024 tiles
    gemm_wmma_f32<B2/16, HID/16, N_FILTER, HID, HID, true>
        <<<(T1 + 3) / 4, 128, 0, stream>>>(feats, W1, b1, h1);
    gemm_wmma_f32<B2/16, HID/16, HID, HID, HID, true>
        <<<(T1 + 3) / 4, 128, 0, stream>>>(h1, W2, b2, h2);
    constexpr int T3 = (B2 / 16) * 1;                    // 128 tiles
    gemm_wmma_f32<B2/16, 1, HID, N_CLASS, 16, false>
        <<<(T3 + 3) / 4, 128, 0, stream>>>(h2, Wl, bl, logits);

    // 3) softmax -> d_out (flat (2048,10), same memory order as reference reshape)
    softmax10<<<(B2 + 255) / 256, 256, 0, stream>>>(logits, out);
}